// SSM_3126736191827
// MI455X (gfx1250) — compile-verified
//
#include <hip/hip_runtime.h>

typedef float v2f __attribute__((ext_vector_type(2)));
typedef float v8f __attribute__((ext_vector_type(8)));

#define BATCH  4
#define SEQ    2048
#define DDIM   768
#define NSTATE 16
#define ROWS   (BATCH * SEQ)   // 8192 rows of x viewed as (ROWS, DDIM)

// ---------------------------------------------------------------------------
// Kernel 1: BC = x @ W_bc^T + b_bc   -> (ROWS, 32), via V_WMMA_F32_16X16X4_F32
// One wave per 16-row M-tile; two f32 accumulators cover the 32 output cols.
// A-frag (16x4 f32, 2 VGPRs): lane holds M=lane&15, K = 2*(lane>>4)+{0,1}
// B-frag (4x16 f32, 2 VGPRs): lane holds N=lane&15, K = 2*(lane>>4)+{0,1}
// D (16x16 f32, 8 VGPRs):     vgpr r -> M = r + 8*(lane>>4), N = lane&15
// ---------------------------------------------------------------------------
__global__ __launch_bounds__(256)
void bc_gemm_wmma(const float* __restrict__ x, const float* __restrict__ Wbc,
                  const float* __restrict__ bbc, float* __restrict__ BC) {
  const int lane   = threadIdx.x & 31;
  const int warp   = threadIdx.x >> 5;
  const int tile_m = blockIdx.x * 8 + warp;          // 512 tiles total
  const int mrow   = lane & 15;
  const int khalf  = (lane >> 4) * 2;
  const int row    = tile_m * 16 + mrow;

  v8f acc0 = {0.f, 0.f, 0.f, 0.f, 0.f, 0.f, 0.f, 0.f};
  v8f acc1 = acc0;

  const float2* xr  = reinterpret_cast<const float2*>(x   + (size_t)row        * DDIM);
  const float2* w0r = reinterpret_cast<const float2*>(Wbc + (size_t)mrow       * DDIM);
  const float2* w1r = reinterpret_cast<const float2*>(Wbc + (size_t)(mrow + 16) * DDIM);

  for (int k = 0; k < DDIM; k += 4) {
    const int i2 = (k + khalf) >> 1;               // float2 index (8B aligned)
    float2 af  = xr[i2];
    float2 b0f = w0r[i2];
    float2 b1f = w1r[i2];
    v2f a;  a[0]  = af.x;  a[1]  = af.y;
    v2f b0; b0[0] = b0f.x; b0[1] = b0f.y;
    v2f b1; b1[0] = b1f.x; b1[1] = b1f.y;
    acc0 = __builtin_amdgcn_wmma_f32_16x16x4_f32(false, a, false, b0,
                                                 (short)0, acc0, false, false);
    acc1 = __builtin_amdgcn_wmma_f32_16x16x4_f32(false, a, false, b1,
                                                 (short)0, acc1, false, false);
  }

  const int ncol   = lane & 15;
  const int rowoff = (lane >> 4) * 8;
  const float bias0 = bbc[ncol];
  const float bias1 = bbc[16 + ncol];
#pragma unroll
  for (int r = 0; r < 8; ++r) {
    const int m = tile_m * 16 + rowoff + r;
    BC[(size_t)m * 32 + ncol]      = acc0[r] + bias0;
    BC[(size_t)m * 32 + 16 + ncol] = acc1[r] + bias1;
  }
}

// ---------------------------------------------------------------------------
// Kernel 2: s1[row] = dot(x[row,:], W_1) + b_1   (one wave per row)
// ---------------------------------------------------------------------------
__global__ __launch_bounds__(256)
void s1_dot(const float* __restrict__ x, const float* __restrict__ W1,
            const float* __restrict__ b1, float* __restrict__ s1) {
  const int lane = threadIdx.x & 31;
  const int row  = blockIdx.x * 8 + (threadIdx.x >> 5);
  const float* xr = x + (size_t)row * DDIM;
  float sum = 0.f;
  for (int k = lane; k < DDIM; k += 32) sum += xr[k] * W1[k];
#pragma unroll
  for (int m = 16; m > 0; m >>= 1) sum += __shfl_xor(sum, m, 32);
  if (lane == 0) s1[row] = sum + b1[0];
}

// ---------------------------------------------------------------------------
// Kernel 3: sequential selective scan over L.
// Lane layout (wave32): lane = n + 16*j ; wave covers d-channels {2p, 2p+1}.
// Per step: delta = softplus(s1*W_d + b_d); h = exp(delta*A)*h + delta*B*x;
//           y[d] = sum_n h*C[n]  (4-step xor-shuffle reduce within 16 lanes).
// Grid: B * D/2 = 1536 independent waves.
// ---------------------------------------------------------------------------
__global__ __launch_bounds__(32)
void ssm_scan(const float* __restrict__ x, const float* __restrict__ BC,
              const float* __restrict__ s1, const float* __restrict__ A_log,
              const float* __restrict__ W_d, const float* __restrict__ b_d,
              float* __restrict__ y) {
  const int lane = threadIdx.x;        // 0..31
  const int n    = lane & 15;
  const int j    = lane >> 4;
  const int idx  = blockIdx.x;         // 0 .. B*D/2 - 1
  const int b    = idx / (DDIM / 2);
  const int d    = (idx % (DDIM / 2)) * 2 + j;

  const float a  = -__expf(A_log[(size_t)d * NSTATE + n]);  // A[d,n]
  const float wd = W_d[d];
  const float bd = b_d[d];

  const float* xb  = x  + (size_t)b * SEQ * DDIM + d;
  const float* BCb = BC + (size_t)b * SEQ * 32;
  const float* s1b = s1 + (size_t)b * SEQ;
  float*       yb  = y  + (size_t)b * SEQ * DDIM + d;

  float h = 0.f;
  for (int l = 0; l < SEQ; ++l) {
    const float bn = BCb[l * 32 + n];
    const float cn = BCb[l * 32 + 16 + n];
    const float xd = xb[(size_t)l * DDIM];   // broadcast within 16-lane group
    const float s  = s1b[l];                 // broadcast within wave
    const float z  = fmaf(s, wd, bd);
    const float e  = __expf(z);
    const float delta = (z > 20.f) ? z : __logf(1.f + e);   // softplus
    const float dA = __expf(delta * a);
    h = fmaf(dA, h, delta * bn * xd);
    float p = h * cn;
    p += __shfl_xor(p, 1, 32);
    p += __shfl_xor(p, 2, 32);
    p += __shfl_xor(p, 4, 32);
    p += __shfl_xor(p, 8, 32);               // sum over n within each half
    if (n == 0) yb[(size_t)l * DDIM] = p;
  }
}

// ---------------------------------------------------------------------------
extern "C" void kernel_launch(void* const* d_in, const int* in_sizes, int n_in,
                              void* d_out, int out_size, void* d_ws, size_t ws_size,
                              hipStream_t stream) {
  const float* x     = (const float*)d_in[0];
  const float* A_log = (const float*)d_in[1];
  const float* W_bc  = (const float*)d_in[2];
  const float* b_bc  = (const float*)d_in[3];
  const float* W_1   = (const float*)d_in[4];
  const float* b_1   = (const float*)d_in[5];
  const float* W_d   = (const float*)d_in[6];
  const float* b_d   = (const float*)d_in[7];
  float* y = (float*)d_out;

  float* BC = (float*)d_ws;                 // ROWS*32 floats = 1 MB
  float* s1 = BC + (size_t)ROWS * 32;       // ROWS floats   = 32 KB

  // 512 M-tiles, 8 waves (tiles) per 256-thread block -> 64 blocks
  bc_gemm_wmma<<<ROWS / 16 / 8, 256, 0, stream>>>(x, W_bc, b_bc, BC);
  // 8 rows per 256-thread block
  s1_dot<<<ROWS / 8, 256, 0, stream>>>(x, W_1, b_1, s1);
  // 1536 single-wave blocks: maximal independent recurrence channels
  ssm_scan<<<BATCH * DDIM / 2, 32, 0, stream>>>(x, BC, s1, A_log, W_d, b_d, y);
}